// OptNet_11690900979744
// MI455X (gfx1250) — compile-verified
//
#include <hip/hip_runtime.h>
#include <hip/hip_bf16.h>
#include <math.h>
#include <stdint.h>

typedef float v2f __attribute__((ext_vector_type(2)));
typedef float v8f __attribute__((ext_vector_type(8)));

#define B_SZ   8192
#define NFEAT  784
#define NHID   2048
#define NCLS   10
#define NINEQ  200
#define IPM_IT 25
#define SIGMA  0.1
#define EPSQ   1e-4
#define BIGV   1e10

// ---- ASYNCcnt waits (CDNA5): builtin when declared, SOPP asm otherwise ----
__device__ __forceinline__ void async_wait_le3() {
#if __has_builtin(__builtin_amdgcn_s_wait_asynccnt)
  __builtin_amdgcn_s_wait_asynccnt(3);
#else
  asm volatile("s_wait_asynccnt 0x3" ::: "memory");
#endif
}
__device__ __forceinline__ void async_wait_le0() {
#if __has_builtin(__builtin_amdgcn_s_wait_asynccnt)
  __builtin_amdgcn_s_wait_asynccnt(0);
#else
  asm volatile("s_wait_asynccnt 0x0" ::: "memory");
#endif
}

// Async global->LDS 16B copy, GVS mode: saddr = uniform base, vaddr = 32-bit
// per-lane byte offset, vdst = LDS byte address (low 32 bits of generic ptr).
__device__ __forceinline__ void async_copy_b128(const void* lds_generic,
                                                const float* gbase,
                                                uint32_t byte_off) {
  uint32_t lds_addr = (uint32_t)(uintptr_t)lds_generic;
  asm volatile("global_load_async_to_lds_b128 %0, %1, %2"
               :: "v"(lds_addr), "v"(byte_off), "s"(gbase)
               : "memory");
}

// ---------------------------------------------------------------------------
// Kernel 1: prep Q = tril(L) tril(L)^T + eps*I  (f64 10x10) and h = G z0 + s0
// ---------------------------------------------------------------------------
__global__ void prep_kernel(const float* __restrict__ L, const float* __restrict__ G,
                            const float* __restrict__ z0, const float* __restrict__ s0,
                            double* __restrict__ Qd, double* __restrict__ hd) {
  int t = threadIdx.x;
  if (t < NCLS * NCLS) {
    int i = t / NCLS, j = t % NCLS;
    int km = i < j ? i : j;
    double acc = 0.0;
    for (int k = 0; k <= km; ++k)
      acc += (double)L[i * NCLS + k] * (double)L[j * NCLS + k];
    if (i == j) acc += EPSQ;
    Qd[t] = acc;
  }
  if (t < NINEQ) {
    double acc = (double)s0[t];
    for (int j = 0; j < NCLS; ++j)
      acc += (double)G[t * NCLS + j] * (double)z0[j];
    hd[t] = acc;
  }
}

// ---------------------------------------------------------------------------
// Kernel 2: h1 = relu(x @ W1^T + b1)   M=8192 K=784 N=2048, fp32 WMMA 16x16x4
// Block = 256 threads (8 waves). Block tile 128(M) x 64(N); wave tile 16x64.
// Double-buffered LDS staged with async global->LDS copies (ASYNCcnt).
// ---------------------------------------------------------------------------
#define TM 128
#define TN 64
#define TK 16
#define LDSTR 20          // padded row stride: 80B rows -> 16B aligned, conflict-free
#define XSTAGE (TM * LDSTR)
#define WSTAGE (TN * LDSTR)
#define NSTAGE (NFEAT / TK)   // 49, exact

__global__ __launch_bounds__(256) void gemm1_kernel(const float* __restrict__ x,
                                                    const float* __restrict__ W1,
                                                    const float* __restrict__ b1,
                                                    float* __restrict__ h1) {
  __shared__ float xs[2 * XSTAGE];
  __shared__ float wsd[2 * WSTAGE];
  const int tid  = threadIdx.x;
  const int wave = tid >> 5;
  const int lane = tid & 31;
  const int l16  = lane & 15;
  const int lh   = lane >> 4;           // 0/1 -> K half
  const int mBase = blockIdx.x * TM;
  const int nBase = blockIdx.y * TN;

  // per-thread staging coordinates (3 x b128 per stage per thread)
  const int xr0 = (tid * 2) >> 2,       xc0 = ((tid * 2) & 3) << 2;
  const int xr1 = (tid * 2 + 1) >> 2,   xc1 = ((tid * 2 + 1) & 3) << 2;
  const int wr  = tid >> 2,             wc  = (tid & 3) << 2;

  auto prefetch = [&](int ks, int buf) {
    async_copy_b128(xs + buf * XSTAGE + xr0 * LDSTR + xc0, x,
                    (uint32_t)(((mBase + xr0) * NFEAT + ks + xc0) * sizeof(float)));
    async_copy_b128(xs + buf * XSTAGE + xr1 * LDSTR + xc1, x,
                    (uint32_t)(((mBase + xr1) * NFEAT + ks + xc1) * sizeof(float)));
    async_copy_b128(wsd + buf * WSTAGE + wr * LDSTR + wc, W1,
                    (uint32_t)(((nBase + wr) * NFEAT + ks + wc) * sizeof(float)));
  };

  v8f acc[4];
#pragma unroll
  for (int j = 0; j < 4; ++j) acc[j] = (v8f){0.f,0.f,0.f,0.f,0.f,0.f,0.f,0.f};

  prefetch(0, 0);
  for (int st = 0; st < NSTAGE; ++st) {
    const int buf = st & 1;
    if (st + 1 < NSTAGE) {
      prefetch((st + 1) * TK, buf ^ 1);  // overlap next stage's HBM traffic
      async_wait_le3();                  // this stage's 3 copies have landed
    } else {
      async_wait_le0();
    }
    __syncthreads();                     // all waves' copies visible in LDS

    const float* xb = xs + buf * XSTAGE;
    const float* wb = wsd + buf * WSTAGE;
#pragma unroll
    for (int kk = 0; kk < TK; kk += 4) {
      // A fragment (16x4 f32): lane l16 = M row, lh selects K pair {0,1}/{2,3}
      v2f a = *(const v2f*)(xb + (wave * 16 + l16) * LDSTR + kk + lh * 2);
#pragma unroll
      for (int j = 0; j < 4; ++j) {
        // B fragment (4x16): l16 = N col of W1^T tile = W1 row
        v2f b = *(const v2f*)(wb + (j * 16 + l16) * LDSTR + kk + lh * 2);
        acc[j] = __builtin_amdgcn_wmma_f32_16x16x4_f32(false, a, false, b,
                                                       (short)0, acc[j], false, false);
      }
    }
    __syncthreads();                     // buffer free before it is refilled
  }

  // epilogue: bias + relu + store (C/D layout: M = v + lh*8, N = l16)
#pragma unroll
  for (int j = 0; j < 4; ++j) {
    int n = nBase + j * 16 + l16;
    float bias = b1[n];
#pragma unroll
    for (int v = 0; v < 8; ++v) {
      int m = mBase + wave * 16 + v + lh * 8;
      float val = acc[j][v] + bias;
      h1[(size_t)m * NHID + n] = val > 0.f ? val : 0.f;
    }
  }
}

// ---------------------------------------------------------------------------
// Kernel 3: p = relu(h1 @ W2^T + b2) as fp64. N=10 padded to one 16-wide tile.
// ---------------------------------------------------------------------------
__global__ __launch_bounds__(256) void gemm2_kernel(const float* __restrict__ h1,
                                                    const float* __restrict__ W2,
                                                    const float* __restrict__ b2,
                                                    double* __restrict__ p) {
  const int tid  = threadIdx.x;
  const int wave = tid >> 5;
  const int lane = tid & 31;
  const int l16  = lane & 15;
  const int lh   = lane >> 4;
  const int mBase = (blockIdx.x * 8 + wave) * 16;

  v8f acc = (v8f){0.f,0.f,0.f,0.f,0.f,0.f,0.f,0.f};
  const float* arow = h1 + (size_t)(mBase + l16) * NHID;
  const bool nvalid = (l16 < NCLS);
  const float* brow = W2 + (size_t)(nvalid ? l16 : 0) * NHID;
  for (int k = 0; k < NHID; k += 4) {
    v2f a = *(const v2f*)(arow + k + lh * 2);
    v2f b = nvalid ? *(const v2f*)(brow + k + lh * 2) : (v2f){0.f, 0.f};
    acc = __builtin_amdgcn_wmma_f32_16x16x4_f32(false, a, false, b,
                                                (short)0, acc, false, false);
  }
  if (nvalid) {
    float bias = b2[l16];
#pragma unroll
    for (int v = 0; v < 8; ++v) {
      int m = mBase + v + lh * 8;
      float val = acc[v] + bias;
      p[(size_t)m * NCLS + l16] = (double)(val > 0.f ? val : 0.f);
    }
  }
}

// ---------------------------------------------------------------------------
// Kernel 4: s = lam = 1
// ---------------------------------------------------------------------------
__global__ void init_slam(double* __restrict__ s, double* __restrict__ lam, int n) {
  int i = blockIdx.x * blockDim.x + threadIdx.x;
  if (i < n) { s[i] = 1.0; lam[i] = 1.0; }
}

// ---------------------------------------------------------------------------
// Kernel 5: fp64 primal-dual IPM, one thread per batch QP (n=10, m=200),
// fused log_softmax. s/lam in [m][batch] layout -> coalesced, L2-resident.
// ---------------------------------------------------------------------------
__device__ __forceinline__ constexpr int UT(int i, int j) {  // packed upper tri, i<=j
  return i * NCLS - (i * (i + 1)) / 2 + j;
}

__global__ __launch_bounds__(128) void ipm_kernel(const double* __restrict__ Qd,
                                                  const double* __restrict__ hd,
                                                  const float* __restrict__ G,
                                                  const double* __restrict__ p,
                                                  double* __restrict__ s_ws,
                                                  double* __restrict__ lam_ws,
                                                  float* __restrict__ out) {
  __shared__ float  Gs[NINEQ * NCLS];
  __shared__ double Qs[NCLS * NCLS];
  __shared__ double hs[NINEQ];
  for (int i = threadIdx.x; i < NINEQ * NCLS; i += blockDim.x) Gs[i] = G[i];
  for (int i = threadIdx.x; i < NCLS * NCLS; i += blockDim.x) Qs[i] = Qd[i];
  for (int i = threadIdx.x; i < NINEQ; i += blockDim.x) hs[i] = hd[i];
  __syncthreads();

  const int b = blockIdx.x * blockDim.x + threadIdx.x;
  double z[NCLS], pb[NCLS];
#pragma unroll
  for (int j = 0; j < NCLS; ++j) { z[j] = 0.0; pb[j] = p[(size_t)b * NCLS + j]; }

  for (int it = 0; it < IPM_IT; ++it) {
    // ---- pass 1: mu = SIGMA * mean(s*lam)
    double sl = 0.0;
    for (int m = 0; m < NINEQ; ++m)
      sl += s_ws[(size_t)m * B_SZ + b] * lam_ws[(size_t)m * B_SZ + b];
    const double mu = SIGMA * sl / (double)NINEQ;

    // ---- pass 2: accumulate H = Q + G^T D G, t1 = lam@G, t2 = (D*rp + rc/s)@G
    double t1[NCLS], t2[NCLS], H[55];
#pragma unroll
    for (int j = 0; j < NCLS; ++j) { t1[j] = 0.0; t2[j] = 0.0; }
#pragma unroll
    for (int i = 0; i < NCLS; ++i)
#pragma unroll
      for (int j = i; j < NCLS; ++j) H[UT(i, j)] = Qs[i * NCLS + j];

    for (int m = 0; m < NINEQ; ++m) {
      double sm = s_ws[(size_t)m * B_SZ + b];
      double lm = lam_ws[(size_t)m * B_SZ + b];
      double g[NCLS];
#pragma unroll
      for (int j = 0; j < NCLS; ++j) g[j] = (double)Gs[m * NCLS + j];
      double gz = 0.0;
#pragma unroll
      for (int j = 0; j < NCLS; ++j) gz += g[j] * z[j];
      double rp  = gz + sm - hs[m];
      double Dm  = lm / sm;
      double rcs = mu / sm - lm;        // r_c / s
      double cm  = Dm * rp + rcs;
#pragma unroll
      for (int j = 0; j < NCLS; ++j) { t1[j] += lm * g[j]; t2[j] += cm * g[j]; }
#pragma unroll
      for (int i = 0; i < NCLS; ++i) {
        double dg = Dm * g[i];
#pragma unroll
        for (int j = i; j < NCLS; ++j) H[UT(i, j)] += dg * g[j];
      }
    }

    // rhs = -(zQ + p + lam@G + (D*rp + rc/s)@G)
    double rhs[NCLS];
#pragma unroll
    for (int j = 0; j < NCLS; ++j) {
      double zq = 0.0;
#pragma unroll
      for (int i = 0; i < NCLS; ++i) zq += z[i] * Qs[i * NCLS + j];
      rhs[j] = -(zq + pb[j] + t1[j] + t2[j]);
    }

    // ---- packed Cholesky H = U^T U (SPD), then two triangular solves
#pragma unroll
    for (int k = 0; k < NCLS; ++k) {
      double d = H[UT(k, k)];
#pragma unroll
      for (int t = 0; t < NCLS; ++t)
        if (t < k) { double u = H[UT(t, k)]; d -= u * u; }
      d = sqrt(d);
      H[UT(k, k)] = d;
      double inv = 1.0 / d;
#pragma unroll
      for (int j = 0; j < NCLS; ++j)
        if (j > k) {
          double v = H[UT(k, j)];
#pragma unroll
          for (int t = 0; t < NCLS; ++t)
            if (t < k) v -= H[UT(t, k)] * H[UT(t, j)];
          H[UT(k, j)] = v * inv;
        }
    }
    double y[NCLS], dz[NCLS];
#pragma unroll
    for (int k = 0; k < NCLS; ++k) {
      double v = rhs[k];
#pragma unroll
      for (int t = 0; t < NCLS; ++t)
        if (t < k) v -= H[UT(t, k)] * y[t];
      y[k] = v / H[UT(k, k)];
    }
#pragma unroll
    for (int k = NCLS - 1; k >= 0; --k) {
      double v = y[k];
#pragma unroll
      for (int t = 0; t < NCLS; ++t)
        if (t > k) v -= H[UT(k, t)] * dz[t];
      dz[k] = v / H[UT(k, k)];
    }

    // ---- pass 3: step length
    double a_s = BIGV, a_l = BIGV;
    for (int m = 0; m < NINEQ; ++m) {
      double sm = s_ws[(size_t)m * B_SZ + b];
      double lm = lam_ws[(size_t)m * B_SZ + b];
      double g[NCLS];
#pragma unroll
      for (int j = 0; j < NCLS; ++j) g[j] = (double)Gs[m * NCLS + j];
      double gz = 0.0, gdz = 0.0;
#pragma unroll
      for (int j = 0; j < NCLS; ++j) { gz += g[j] * z[j]; gdz += g[j] * dz[j]; }
      double rp   = gz + sm - hs[m];
      double Dm   = lm / sm;
      double rcs  = mu / sm - lm;
      double dlam = Dm * (gdz + rp) + rcs;
      double dsm  = (mu - sm * lm - sm * dlam) / lm;
      if (dsm  < 0.0) a_s = fmin(a_s, -sm / dsm);
      if (dlam < 0.0) a_l = fmin(a_l, -lm / dlam);
    }
    const double a = fmin(1.0, 0.99 * fmin(a_s, a_l));

    // ---- pass 4: update s, lam, z
    for (int m = 0; m < NINEQ; ++m) {
      double sm = s_ws[(size_t)m * B_SZ + b];
      double lm = lam_ws[(size_t)m * B_SZ + b];
      double g[NCLS];
#pragma unroll
      for (int j = 0; j < NCLS; ++j) g[j] = (double)Gs[m * NCLS + j];
      double gz = 0.0, gdz = 0.0;
#pragma unroll
      for (int j = 0; j < NCLS; ++j) { gz += g[j] * z[j]; gdz += g[j] * dz[j]; }
      double rp   = gz + sm - hs[m];
      double Dm   = lm / sm;
      double rcs  = mu / sm - lm;
      double dlam = Dm * (gdz + rp) + rcs;
      double dsm  = (mu - sm * lm - sm * dlam) / lm;
      s_ws[(size_t)m * B_SZ + b]   = sm + a * dsm;
      lam_ws[(size_t)m * B_SZ + b] = lm + a * dlam;
    }
#pragma unroll
    for (int j = 0; j < NCLS; ++j) z[j] += a * dz[j];
  }

  // fused log_softmax in fp32
  float zf[NCLS], mx = -3.402823466e38f;
#pragma unroll
  for (int j = 0; j < NCLS; ++j) { zf[j] = (float)z[j]; mx = fmaxf(mx, zf[j]); }
  float se = 0.f;
#pragma unroll
  for (int j = 0; j < NCLS; ++j) se += __expf(zf[j] - mx);
  const float lse = __logf(se) + mx;
#pragma unroll
  for (int j = 0; j < NCLS; ++j) out[b * NCLS + j] = zf[j] - lse;
}

// ---------------------------------------------------------------------------
extern "C" void kernel_launch(void* const* d_in, const int* in_sizes, int n_in,
                              void* d_out, int out_size, void* d_ws, size_t ws_size,
                              hipStream_t stream) {
  const float* x  = (const float*)d_in[0];
  const float* W1 = (const float*)d_in[1];
  const float* b1 = (const float*)d_in[2];
  const float* W2 = (const float*)d_in[3];
  const float* b2 = (const float*)d_in[4];
  const float* L  = (const float*)d_in[5];
  const float* G  = (const float*)d_in[6];
  const float* z0 = (const float*)d_in[7];
  const float* s0 = (const float*)d_in[8];

  char* ws = (char*)d_ws;
  size_t off = 0;
  float*  h1     = (float*) (ws + off); off += (size_t)B_SZ * NHID * sizeof(float);   // 64 MB
  double* p      = (double*)(ws + off); off += (size_t)B_SZ * NCLS * sizeof(double);
  double* s_ws   = (double*)(ws + off); off += (size_t)NINEQ * B_SZ * sizeof(double);
  double* lam_ws = (double*)(ws + off); off += (size_t)NINEQ * B_SZ * sizeof(double);
  double* Qd     = (double*)(ws + off); off += (size_t)NCLS * NCLS * sizeof(double);
  double* hd     = (double*)(ws + off); off += (size_t)NINEQ * sizeof(double);

  prep_kernel<<<1, 256, 0, stream>>>(L, G, z0, s0, Qd, hd);

  dim3 g1(B_SZ / TM, NHID / TN);
  gemm1_kernel<<<g1, 256, 0, stream>>>(x, W1, b1, h1);

  gemm2_kernel<<<B_SZ / 128, 256, 0, stream>>>(h1, W2, b2, p);

  int nsl = B_SZ * NINEQ;
  init_slam<<<(nsl + 255) / 256, 256, 0, stream>>>(s_ws, lam_ws, nsl);

  ipm_kernel<<<B_SZ / 128, 128, 0, stream>>>(Qd, hd, G, p, s_ws, lam_ws, (float*)d_out);
}